// GraphAttnGATAggregationModule_75050258530504
// MI455X (gfx1250) — compile-verified
//
#include <hip/hip_runtime.h>
#include <math.h>

#define NEG_SLOPE 0.2f

typedef __attribute__((ext_vector_type(2))) float v2f;
typedef __attribute__((ext_vector_type(8))) float v8f;

// ---------------------------------------------------------------------------
// Kernel 1: s[N,8] = [ x @ W_u + b_u  |  x @ W_v ]  via V_WMMA_F32_16X16X4_F32
// One wave32 per 16-node tile. Combined weights staged as 128x16 block in LDS
// (cols 0-3 = W_u, cols 4-7 = W_v, cols 8-15 = 0 padding).
// ---------------------------------------------------------------------------
__global__ __launch_bounds__(32) void gat_scores_wmma(
    const float* __restrict__ x, const float* __restrict__ W_u,
    const float* __restrict__ b_u, const float* __restrict__ W_v,
    float* __restrict__ s, int n_nodes) {
  const int tile = blockIdx.x;
  const int lane = threadIdx.x;      // 0..31
  const int half = lane >> 4;        // 0: lanes 0-15, 1: lanes 16-31
  const int l15  = lane & 15;

  __shared__ float Wc[128 * 16];
  for (int i = lane; i < 128 * 16; i += 32) {
    const int k = i >> 4, c = i & 15;
    float w = 0.0f;
    if (c < 4)      w = W_u[k * 4 + c];
    else if (c < 8) w = W_v[k * 4 + (c - 4)];
    Wc[i] = w;
  }
  __syncthreads();

  int node = tile * 16 + l15;
  const int nodeC = node < n_nodes ? node : n_nodes - 1;   // keep EXEC all-1s
  const float* __restrict__ xrow = x + (size_t)nodeC * 128;

  v8f acc = {};
  #pragma unroll 4
  for (int k = 0; k < 128; k += 4) {
    // A 16x4 f32 fragment: VGPR0 = {K=k | K=k+2}, VGPR1 = {K=k+1 | K=k+3}
    const int ka = k + half * 2;
    v2f a; a.x = xrow[ka];            a.y = xrow[ka + 1];
    // B 4x16 f32 fragment, mirrored row striping across lanes
    v2f b; b.x = Wc[ka * 16 + l15];   b.y = Wc[(ka + 1) * 16 + l15];
    acc = __builtin_amdgcn_wmma_f32_16x16x4_f32(
        /*neg_a=*/false, a, /*neg_b=*/false, b,
        /*c_mod=*/(short)0, acc, /*reuse_a=*/false, /*reuse_b=*/false);
  }

  // C/D layout: VGPR r -> lanes 0-15: M=r, N=lane; lanes 16-31: M=r+8, N=lane-16
  const int ncol = l15;
  if (ncol < 8) {
    const float bias = (ncol < 4) ? b_u[ncol] : 0.0f;
    #pragma unroll
    for (int r = 0; r < 8; ++r) {
      const int mrow = r + half * 8;
      const int onode = tile * 16 + mrow;
      if (onode < n_nodes) s[(size_t)onode * 8 + ncol] = acc[r] + bias;
    }
  }
}

// ---------------------------------------------------------------------------
// Kernel 2a: m = -inf, denom = 0
// ---------------------------------------------------------------------------
__global__ void gat_init_md(float* __restrict__ m, float* __restrict__ denom,
                            int n4) {
  const int i = blockIdx.x * blockDim.x + threadIdx.x;
  if (i < n4) { m[i] = -INFINITY; denom[i] = 0.0f; }
}

// ---------------------------------------------------------------------------
// Kernel 2b: out[n, 0:128] = x[n]; out[n, 128:256] = 0
// ---------------------------------------------------------------------------
__global__ void gat_init_out(const float* __restrict__ x,
                             float* __restrict__ out, long long total) {
  const long long i = (long long)blockIdx.x * blockDim.x + threadIdx.x;
  if (i >= total) return;
  const long long node = i >> 8;
  const int d = (int)(i & 255);
  out[i] = (d < 128) ? x[node * 128 + d] : 0.0f;
}

// ---------------------------------------------------------------------------
// Float atomic max via monotone int/uint punning (valid for all finite + -inf)
// ---------------------------------------------------------------------------
__device__ __forceinline__ void atomicMaxFloat(float* addr, float val) {
  if (val >= 0.0f) atomicMax((int*)addr, __float_as_int(val));
  else             atomicMin((unsigned int*)addr, __float_as_uint(val));
}

__device__ __forceinline__ float lrelu(float v) {
  return v >= 0.0f ? v : NEG_SLOPE * v;
}

// ---------------------------------------------------------------------------
// Kernel 3: segment max of edge scores over dst
// ---------------------------------------------------------------------------
__global__ void gat_edge_max(const float* __restrict__ s,
                             const int* __restrict__ src,
                             const int* __restrict__ dst,
                             float* __restrict__ m, int E) {
  const int e = blockIdx.x * blockDim.x + threadIdx.x;
  if (e >= E) return;
  const int u = src[e], v = dst[e];
  const float4 su = *(const float4*)(s + (size_t)u * 8);
  const float4 sv = *(const float4*)(s + (size_t)v * 8 + 4);
  float* mv = m + (size_t)v * 4;
  atomicMaxFloat(mv + 0, lrelu(su.x + sv.x));
  atomicMaxFloat(mv + 1, lrelu(su.y + sv.y));
  atomicMaxFloat(mv + 2, lrelu(su.z + sv.z));
  atomicMaxFloat(mv + 3, lrelu(su.w + sv.w));
}

// ---------------------------------------------------------------------------
// Kernel 4: denom = segment_sum(exp(e - m[dst]))
// ---------------------------------------------------------------------------
__global__ void gat_edge_denom(const float* __restrict__ s,
                               const int* __restrict__ src,
                               const int* __restrict__ dst,
                               const float* __restrict__ m,
                               float* __restrict__ denom, int E) {
  const int e = blockIdx.x * blockDim.x + threadIdx.x;
  if (e >= E) return;
  const int u = src[e], v = dst[e];
  const float4 su = *(const float4*)(s + (size_t)u * 8);
  const float4 sv = *(const float4*)(s + (size_t)v * 8 + 4);
  const float4 mv = *(const float4*)(m + (size_t)v * 4);
  float* dv = denom + (size_t)v * 4;
  atomicAdd(dv + 0, __expf(lrelu(su.x + sv.x) - mv.x));
  atomicAdd(dv + 1, __expf(lrelu(su.y + sv.y) - mv.y));
  atomicAdd(dv + 2, __expf(lrelu(su.z + sv.z) - mv.z));
  atomicAdd(dv + 3, __expf(lrelu(su.w + sv.w) - mv.w));
}

// ---------------------------------------------------------------------------
// Kernel 5: wave32 per edge. Lanes 0-3 own the 4 head probabilities
// (computed redundantly in all lanes to keep EXEC uniform, broadcast by shfl).
// Each lane gathers float4 of x[src] (512B coalesced per wave) and scatter-
// adds into out[dst, 128:256]. Head of output dim d is (d & 3) per the
// reference reshape(N, HEAD_DIM, NUM_HEADS).
// ---------------------------------------------------------------------------
__global__ void gat_edge_agg(const float* __restrict__ x,
                             const float* __restrict__ s,
                             const int* __restrict__ src,
                             const int* __restrict__ dst,
                             const float* __restrict__ m,
                             const float* __restrict__ denom,
                             float* __restrict__ out, int E) {
  const long long gid = (long long)blockIdx.x * blockDim.x + threadIdx.x;
  const int e = (int)(gid >> 5);
  const int lane = (int)(gid & 31);
  if (e >= E) return;                    // uniform per wave
  const int u = src[e], v = dst[e];

  const int h = lane & 3;
  const float ee = lrelu(s[(size_t)u * 8 + h] + s[(size_t)v * 8 + 4 + h]);
  const float p = __expf(ee - m[(size_t)v * 4 + h]) / denom[(size_t)v * 4 + h];
  const float p0 = __shfl(p, 0, 32);
  const float p1 = __shfl(p, 1, 32);
  const float p2 = __shfl(p, 2, 32);
  const float p3 = __shfl(p, 3, 32);

  const float4 xv = *(const float4*)(x + (size_t)u * 128 + lane * 4);
  float* o = out + (size_t)v * 256 + 128 + lane * 4;
  atomicAdd(o + 0, xv.x * p0);
  atomicAdd(o + 1, xv.y * p1);
  atomicAdd(o + 2, xv.z * p2);
  atomicAdd(o + 3, xv.w * p3);
}

extern "C" void kernel_launch(void* const* d_in, const int* in_sizes, int n_in,
                              void* d_out, int out_size, void* d_ws, size_t ws_size,
                              hipStream_t stream) {
  const float* x   = (const float*)d_in[0];
  const int*   src = (const int*)d_in[1];
  const int*   dst = (const int*)d_in[2];
  const float* W_u = (const float*)d_in[3];
  const float* b_u = (const float*)d_in[4];
  const float* W_v = (const float*)d_in[5];
  float* out = (float*)d_out;

  const int n = in_sizes[0] / 128;   // N_NODES
  const int E = in_sizes[1];         // N_EDGES

  // workspace: s[N*8] | m[N*4] | denom[N*4]   (~6.4 MB)
  float* s     = (float*)d_ws;
  float* m     = s + (size_t)n * 8;
  float* denom = m + (size_t)n * 4;

  // 1) node scores via WMMA (one wave per 16-node tile)
  gat_scores_wmma<<<(n + 15) / 16, 32, 0, stream>>>(x, W_u, b_u, W_v, s, n);

  // 2) initialize softmax state and output
  gat_init_md<<<(n * 4 + 255) / 256, 256, 0, stream>>>(m, denom, n * 4);
  const long long total = (long long)n * 256;
  gat_init_out<<<(unsigned)((total + 255) / 256), 256, 0, stream>>>(x, out, total);

  // 3) segment max
  gat_edge_max<<<(E + 255) / 256, 256, 0, stream>>>(s, src, dst, m, E);

  // 4) segment sum of exp
  gat_edge_denom<<<(E + 255) / 256, 256, 0, stream>>>(s, src, dst, m, denom, E);

  // 5) weighted aggregation, one wave32 per edge (8 edges per 256-thread block)
  gat_edge_agg<<<(E + 7) / 8, 256, 0, stream>>>(x, s, src, dst, m, denom, out, E);
}